// EdgeLoss_42374147343038
// MI455X (gfx1250) — compile-verified
//
#include <hip/hip_runtime.h>
#include <stdint.h>

// EdgeLoss on MI455X (gfx1250):
//   loss = mean|sobel_x(gray(sr)-gray(hr))| + mean|sobel_y(gray(sr)-gray(hr))|
// Memory-bound streaming stencil (201 MB in, 1 float out, ~8.6us at 23.3 TB/s).
// Bulk data movement via Tensor Data Mover (tensor_load_to_lds) with
// s_wait_tensorcnt sync; rolling 2-band window per block cuts halo re-reads
// to 1.125x. Deterministic two-level reduction.

namespace {
constexpr int kH = 512, kW = 512, kB = 32;
constexpr int kR = 8;                  // interior rows per band
constexpr int kBR = kR + 2;            // rows incl. +/-1 halo
constexpr int kGP = kW + 2;            // gray-diff LDS pitch (zero border cols)
constexpr int kT = 256;                // threads per block (8 waves @ wave32)
constexpr int kBands = kH / kR;        // 64
constexpr int kSteps = 2;              // consecutive bands per block (rolling)
constexpr int kBPI = kBands / kSteps;  // 32 blocks per image
constexpr int kBlocks = kB * kBPI;     // 1024
constexpr int kGrayPerThr = (kBR * kW) / kT;  // 20 (first band: 10 rows)
constexpr int kNewPerThr  = (kR  * kW) / kT;  // 16 (later bands: 8 new rows)
constexpr int kOutPerThr  = (kR  * kW) / kT;  // 16 outputs / thread / band
}

typedef uint32_t u32x4 __attribute__((ext_vector_type(4)));
typedef int      i32x4 __attribute__((ext_vector_type(4)));
typedef int      i32x8 __attribute__((ext_vector_type(8)));

__device__ __forceinline__ uint32_t lds_off(const void* p) {
  // LDS aperture base occupies bits [63:32]; low 32 bits = wave-relative offset.
  return (uint32_t)(uintptr_t)p;
}

// One TDM 2D tile load: tile_rows x 512 f32 (row pitch 512 elems) from gsrc
// into LDS at byte offset lds_byte. Rows with y >= valid_rows are OOB in dim1
// and written as ZERO by the TDM (free bottom-edge SAME padding).
__device__ __forceinline__ void tdm_load_rows(uint32_t lds_byte,
                                              const float* gsrc,
                                              int tile_rows,
                                              int valid_rows) {
  uint64_t ga = (uint64_t)(uintptr_t)gsrc & 0x01FFFFFFFFFFFFFFull;  // 57-bit VA
  u32x4 g0;
  g0[0] = 1u;                                   // count=1 (valid), user mode
  g0[1] = lds_byte;                             // lds_addr
  g0[2] = (uint32_t)ga;                         // global_addr[31:0]
  g0[3] = (uint32_t)(ga >> 32) | (2u << 30);    // global_addr[56:32] | type=2
  i32x8 g1;
  g1[0] = 0x00020000;                           // data_size=2 (4 bytes)
  g1[1] = (int)((uint32_t)kW << 16);            // tensor_dim0 = 512
  g1[2] = (int)(((uint32_t)valid_rows & 0xFFFFu) << 16);   // tensor_dim1 lo
  g1[3] = (int)(((uint32_t)valid_rows >> 16) |
                ((uint32_t)kW << 16));          // tensor_dim1 hi | tile_dim0=512
  g1[4] = tile_rows;                            // tile_dim1 (tile_dim2 = 0)
  g1[5] = kW;                                   // tensor_dim0_stride = 512
  g1[6] = 0;
  g1[7] = 0;
  i32x4 z4 = {0, 0, 0, 0};                      // groups 2/3: dims >2 unused
#if defined(__clang_major__) && (__clang_major__ >= 23)
  i32x8 z8 = {0, 0, 0, 0, 0, 0, 0, 0};
  __builtin_amdgcn_tensor_load_to_lds(g0, g1, z4, z4, z8, 0);
#else
  __builtin_amdgcn_tensor_load_to_lds(g0, g1, z4, z4, 0);
#endif
}

__global__ __launch_bounds__(kT)
void edge_partial(const float* __restrict__ sr, const float* __restrict__ hr,
                  float* __restrict__ partial) {
  __shared__ float bufA[kBR * kW];   // sr channel rows (20 KB)
  __shared__ float bufB[kBR * kW];   // hr channel rows (20 KB)
  __shared__ float gd[kBR * kGP];    // rolling gray-diff window (20.6 KB)
  __shared__ float red[kT];

  const int tid = threadIdx.x;
  const int img = blockIdx.x / kBPI;
  const int seg = blockIdx.x % kBPI;
  const size_t plane = (size_t)kH * kW;
  const float wgt[3] = {0.2989f, 0.587f, 0.114f};
  const float* sbase = sr + (size_t)img * 3 * plane;
  const float* hbase = hr + (size_t)img * 3 * plane;

  float s = 0.0f;
  float acc[kGrayPerThr];

  for (int t = 0; t < kSteps; ++t) {
    const int band = seg * kSteps + t;
    const int y0 = band * kR;

    if (t == 0) {
      // ---- full 10-row window: y0-1 .. y0+8 ----
      const int y_lo = y0 - 1;
      const int lead = (band == 0) ? 1 : 0;      // top halo off-image
      const int y_start = y_lo + lead;
      const int tile_rows = kBR - lead;
      const int valid_rows = kH - y_start;
      #pragma unroll
      for (int c = 0; c < 3; ++c) {
        if (tid < 32) {  // wave 0 issues DMA (EXEC ignored by TDM)
          tdm_load_rows(lds_off(bufA) + (uint32_t)(lead * kW * 4),
                        sbase + (size_t)c * plane + (size_t)y_start * kW,
                        tile_rows, valid_rows);
          tdm_load_rows(lds_off(bufB) + (uint32_t)(lead * kW * 4),
                        hbase + (size_t)c * plane + (size_t)y_start * kW,
                        tile_rows, valid_rows);
          __builtin_amdgcn_s_wait_tensorcnt(0);
        }
        __syncthreads();
        #pragma unroll
        for (int k = 0; k < kGrayPerThr; ++k) {
          int i = tid + k * kT;
          float v = wgt[c] * (bufA[i] - bufB[i]);
          acc[k] = (c == 0) ? v : (acc[k] + v);
        }
        __syncthreads();
      }
      #pragma unroll
      for (int k = 0; k < kGrayPerThr; ++k) {
        int i = tid + k * kT;
        int r = i >> 9, x = i & (kW - 1);
        gd[r * kGP + x + 1] = ((y_lo + r) >= 0) ? acc[k] : 0.0f;
      }
      if (tid < kBR) {  // border cols: zeroed once, never touched again
        gd[tid * kGP] = 0.0f;
        gd[tid * kGP + kW + 1] = 0.0f;
      }
      __syncthreads();
    } else {
      // ---- rolling step: keep 2 rows, DMA 8 new rows (y0+1 .. y0+8) ----
      __syncthreads();  // previous stencil reads complete before shift
      for (int i = tid; i < 2 * kGP; i += kT)   // rows 8,9 -> rows 0,1
        gd[i] = gd[kR * kGP + i];
      const int y_start = y0 + 1;
      const int valid_rows = kH - y_start;
      #pragma unroll
      for (int c = 0; c < 3; ++c) {
        if (tid < 32) {
          tdm_load_rows(lds_off(bufA),
                        sbase + (size_t)c * plane + (size_t)y_start * kW,
                        kR, valid_rows);
          tdm_load_rows(lds_off(bufB),
                        hbase + (size_t)c * plane + (size_t)y_start * kW,
                        kR, valid_rows);
          __builtin_amdgcn_s_wait_tensorcnt(0);
        }
        __syncthreads();
        #pragma unroll
        for (int k = 0; k < kNewPerThr; ++k) {
          int i = tid + k * kT;
          float v = wgt[c] * (bufA[i] - bufB[i]);
          acc[k] = (c == 0) ? v : (acc[k] + v);
        }
        __syncthreads();
      }
      #pragma unroll
      for (int k = 0; k < kNewPerThr; ++k) {
        int i = tid + k * kT;
        int r = (i >> 9) + 2, x = i & (kW - 1);
        gd[r * kGP + x + 1] = acc[k];
      }
      __syncthreads();
    }

    // ---- Sobel stencil + |.| over rows 1..8 (zero-center kernels: 8 taps) ----
    #pragma unroll
    for (int k = 0; k < kOutPerThr; ++k) {
      int i = tid + k * kT;
      int r = (i >> 9) + 1;
      int x = (i & (kW - 1)) + 1;
      const float* r0 = &gd[(r - 1) * kGP + x];
      const float* r1 = &gd[r * kGP + x];
      const float* r2 = &gd[(r + 1) * kGP + x];
      float a = r0[-1], b = r0[0], c = r0[1];
      float d = r1[-1],            e = r1[1];
      float f = r2[-1], g = r2[0], h = r2[1];
      float gx = (a - c) + 2.0f * (d - e) + (f - h);
      float gy = (a + 2.0f * b + c) - (f + 2.0f * g + h);
      s += fabsf(gx) + fabsf(gy);
    }
  }

  red[tid] = s;
  __syncthreads();
  #pragma unroll
  for (int off = kT / 2; off > 0; off >>= 1) {
    if (tid < off) red[tid] += red[tid + off];
    __syncthreads();
  }
  if (tid == 0) partial[blockIdx.x] = red[0];
}

__global__ __launch_bounds__(kT)
void edge_final(const float* __restrict__ partial, float* __restrict__ out) {
  __shared__ float red[kT];
  float s = 0.0f;
  for (int i = threadIdx.x; i < kBlocks; i += kT) s += partial[i];  // fixed order
  red[threadIdx.x] = s;
  __syncthreads();
  #pragma unroll
  for (int off = kT / 2; off > 0; off >>= 1) {
    if (threadIdx.x < off) red[threadIdx.x] += red[threadIdx.x + off];
    __syncthreads();
  }
  if (threadIdx.x == 0)
    out[0] = red[0] * (1.0f / 8388608.0f);  // / (32*512*512), exact pow2
}

extern "C" void kernel_launch(void* const* d_in, const int* in_sizes, int n_in,
                              void* d_out, int out_size, void* d_ws, size_t ws_size,
                              hipStream_t stream) {
  (void)in_sizes; (void)n_in; (void)out_size; (void)ws_size;
  const float* sr = (const float*)d_in[0];
  const float* hr = (const float*)d_in[1];
  float* partial = (float*)d_ws;  // kBlocks floats = 4 KB
  edge_partial<<<dim3(kBlocks), dim3(kT), 0, stream>>>(sr, hr, partial);
  edge_final<<<1, kT, 0, stream>>>(partial, (float*)d_out);
}